// Layer_63556926046533
// MI455X (gfx1250) — compile-verified
//
#include <hip/hip_runtime.h>

typedef __attribute__((ext_vector_type(2))) float v2f;
typedef __attribute__((ext_vector_type(8))) float v8f;

#define S_LEN 2048
#define D_DIM 768
#define HD 128
#define QH 6
#define KVH 3
#define NSLOT 12        // QH + 2*KVH
#define NOSLOT 6
#define E_ATTN 96
#define E_O 48
#define E_FFN 48
#define FFN_ED 256
#define QKV_STRIDE (NSLOT * HD)   // 1536

__device__ __forceinline__ v8f wmma4(v2f a, v2f b, v8f c) {
  return __builtin_amdgcn_wmma_f32_16x16x4_f32(false, a, false, b, (short)0, c, false, false);
}

// ---- 32x32 register-blocked tile: C[2][2] += A(32xK, rm lda) * B(Kx32, rm ldb)
// 6 loads per 4 WMMA, 4 independent accumulation chains.
__device__ __forceinline__ void wmma_tile_rr_2x2(const float* __restrict__ A, int lda,
                                                 const float* __restrict__ B, int ldb,
                                                 int K, v8f c[2][2], int lane) {
  const int half = lane >> 4, l16 = lane & 15;
  const float* a0 = A + l16 * lda + half * 2;
  const float* a1 = a0 + 16 * lda;
  const float* b0 = B + l16 + (half * 2) * ldb;
  const float* b1 = b0 + 16;
  for (int k0 = 0; k0 < K; k0 += 4) {
    v2f av0 = *(const v2f*)(a0 + k0);
    v2f av1 = *(const v2f*)(a1 + k0);
    v2f bv0, bv1;
    bv0.x = b0[k0 * ldb]; bv0.y = b0[k0 * ldb + ldb];
    bv1.x = b1[k0 * ldb]; bv1.y = b1[k0 * ldb + ldb];
    c[0][0] = wmma4(av0, bv0, c[0][0]);
    c[0][1] = wmma4(av0, bv1, c[0][1]);
    c[1][0] = wmma4(av1, bv0, c[1][0]);
    c[1][1] = wmma4(av1, bv1, c[1][1]);
  }
}

// ---- same with B given transposed: Bt(32xK, rm ldbt), Bt[n][k] == B[k][n]
__device__ __forceinline__ void wmma_tile_rt_2x2(const float* __restrict__ A, int lda,
                                                 const float* __restrict__ Bt, int ldbt,
                                                 int K, v8f c[2][2], int lane) {
  const int half = lane >> 4, l16 = lane & 15;
  const float* a0 = A + l16 * lda + half * 2;
  const float* a1 = a0 + 16 * lda;
  const float* bt0 = Bt + l16 * ldbt + half * 2;
  const float* bt1 = bt0 + 16 * ldbt;
  for (int k0 = 0; k0 < K; k0 += 4) {
    v2f av0 = *(const v2f*)(a0 + k0);
    v2f av1 = *(const v2f*)(a1 + k0);
    v2f bv0 = *(const v2f*)(bt0 + k0);
    v2f bv1 = *(const v2f*)(bt1 + k0);
    c[0][0] = wmma4(av0, bv0, c[0][0]);
    c[0][1] = wmma4(av0, bv1, c[0][1]);
    c[1][0] = wmma4(av1, bv0, c[1][0]);
    c[1][1] = wmma4(av1, bv1, c[1][1]);
  }
}

__device__ __forceinline__ void zero22(v8f c[2][2]) {
  const v8f vz = {0.f, 0.f, 0.f, 0.f, 0.f, 0.f, 0.f, 0.f};
  c[0][0] = vz; c[0][1] = vz; c[1][0] = vz; c[1][1] = vz;
}

// ---------------------------------------------------------------- RMSNorm
__global__ void rmsnorm_kernel(const float* __restrict__ x, const float* __restrict__ w,
                               float* __restrict__ out) {
  const int s = blockIdx.x;
  const float* row = x + s * D_DIM;
  __shared__ float red[8];
  __shared__ float sscale;
  float acc = 0.f;
  for (int i = threadIdx.x; i < D_DIM; i += 256) { float v = row[i]; acc += v * v; }
  #pragma unroll
  for (int off = 16; off > 0; off >>= 1) acc += __shfl_xor(acc, off, 32);
  if ((threadIdx.x & 31) == 0) red[threadIdx.x >> 5] = acc;
  __syncthreads();
  if (threadIdx.x == 0) {
    float t = 0.f;
    #pragma unroll
    for (int i = 0; i < 8; i++) t += red[i];
    sscale = rsqrtf(t / (float)D_DIM + 1e-6f);
  }
  __syncthreads();
  const float sc = sscale;
  for (int i = threadIdx.x; i < D_DIM; i += 256) out[s * D_DIM + i] = row[i] * sc * w[i];
}

// ---------------------------------------------------------------- top-1 routing (attn & o)
template <int RD, int QD, int E>
__global__ void route_top1_kernel(const float* __restrict__ xin, const float* __restrict__ kp,
                                  float* __restrict__ g_out, int* __restrict__ idx_out,
                                  int n_rows, int slots, int row_stride, int slot_stride) {
  __shared__ float skp[(QD + 4) * E];
  for (int i = threadIdx.x; i < (QD + 4) * E; i += 256) skp[i] = kp[i];
  __syncthreads();
  const int r = blockIdx.x * 256 + threadIdx.x;
  if (r >= n_rows) return;
  const int s = r / slots, slot = r % slots;
  const float* q = xin + s * row_stride + slot * slot_stride;
  float qv[QD];
  #pragma unroll
  for (int j = 0; j < QD; j++) qv[j] = q[j];
  float bestv = -3.0e38f; int besti = 0;
  for (int e = 0; e < E; e++) {
    float acc = skp[QD * E + e];           // kp[-4] logit bias
    #pragma unroll
    for (int j = 0; j < RD; j++) acc += qv[j] * skp[j * E + e];
    if (acc > bestv) { bestv = acc; besti = e; }
  }
  float gate = 0.f;
  #pragma unroll
  for (int j = RD; j < QD; j++) gate += qv[j] * skp[j * E + besti];
  const float z = bestv + gate + skp[(QD + 1) * E + besti];
  const float sg = 1.f / (1.f + __expf(-z));
  g_out[r] = sg * skp[(QD + 2) * E + besti] + skp[(QD + 3) * E + besti];
  idx_out[r] = besti;
}

// ---------------------------------------------------------------- per-expert GEMM: all_out[s,e,:] = h @ W_e
__global__ __launch_bounds__(256) void expert_gemm_attn_kernel(const float* __restrict__ h,
                                                               const float* __restrict__ pool,
                                                               float* __restrict__ all_out) {
  const int lane = threadIdx.x & 31;
  const int wid = blockIdx.x * 8 + (threadIdx.x >> 5);
  const int e = wid >> 8;             // 256 32x32-tiles/expert (64 M x 4 N)
  const int t = wid & 255;
  const int mt = t >> 2, nt = t & 3;
  const int m0 = mt * 32, n0 = nt * 32;
  v8f c[2][2];
  zero22(c);
  wmma_tile_rr_2x2(h + (size_t)m0 * D_DIM, D_DIM,
                   pool + (size_t)e * (D_DIM * HD) + n0, HD, D_DIM, c, lane);
  const int half = lane >> 4, l16 = lane & 15;
  float* out = all_out + (size_t)m0 * (E_ATTN * HD) + e * HD + n0 + l16;
  #pragma unroll
  for (int mi = 0; mi < 2; mi++)
    #pragma unroll
    for (int r = 0; r < 8; r++) {
      const int row = mi * 16 + (half ? 8 + r : r);
      #pragma unroll
      for (int ni = 0; ni < 2; ni++)
        out[(size_t)row * (E_ATTN * HD) + ni * 16] = c[mi][ni][r];
    }
}

// ---------------------------------------------------------------- qkv = dense + g*moe
__global__ void qkv_build_kernel(const float* __restrict__ all_out, const float* __restrict__ g,
                                 const int* __restrict__ idx, float* __restrict__ qkv) {
  const int t = blockIdx.x * 256 + threadIdx.x;       // S*12*128
  const int d = t & 127;
  const int slot = (t >> 7) % NSLOT;
  const int s = t / (NSLOT * HD);
  const size_t base = (size_t)s * (E_ATTN * HD);
  const int rr = s * NSLOT + slot;
  qkv[t] = all_out[base + slot * HD + d] + g[rr] * all_out[base + idx[rr] * HD + d];
}

// ---------------------------------------------------------------- RoPE on q (slots 0..5) and k (6..8)
__global__ void rope_kernel(float* __restrict__ qkv) {
  const int t = blockIdx.x * 256 + threadIdx.x;       // S*9*64
  const int i = t & 63;
  const int slot = (t >> 6) % 9;
  const int s = t / (9 * 64);
  float* base = qkv + (size_t)(s * NSLOT + slot) * HD;
  const float inv = __expf(-(float)i * (9.210340371976184f / 64.0f));  // (1/1e4)^(i/64)
  const float f = (float)s * inv;
  const float cf = __cosf(f), sf = __sinf(f);
  const float x1 = base[i], x2 = base[i + 64];
  base[i] = x1 * cf + x2 * sf;
  base[i + 64] = -x1 * sf + x2 * cf;
}

// ---------------------------------------------------------------- flash attention (causal, window >= S)
__global__ __launch_bounds__(256) void flash_attn_kernel(const float* __restrict__ qkv,
                                                         float* __restrict__ xattn) {
  __shared__ __align__(16) float plds[8][16 * 18];
  const int lane = threadIdx.x & 31;
  const int widb = threadIdx.x >> 5;
  const int wid = blockIdx.x * 8 + widb;              // [0, 6*128)
  const int head = wid >> 7;
  const int q0 = (wid & 127) * 16;
  const int kv = head >> 1;
  const float* Q = qkv + head * HD;
  const float* K = qkv + (QH + kv) * HD;
  const float* V = qkv + (QH + KVH + kv) * HD;
  const int half = lane >> 4, l16 = lane & 15;
  float* pw = &plds[widb][0];

  v2f qf[32];
  #pragma unroll
  for (int kk = 0; kk < 32; kk++)
    qf[kk] = *(const v2f*)(Q + (size_t)(q0 + l16) * QKV_STRIDE + kk * 4 + half * 2);

  const v8f vz = {0.f, 0.f, 0.f, 0.f, 0.f, 0.f, 0.f, 0.f};
  v8f o[8];
  float m[8], l[8];
  #pragma unroll
  for (int r = 0; r < 8; r++) { o[r] = vz; m[r] = -3.0e38f; l[r] = 0.f; }

  const float scale = 0.08838834764831845f;           // 1/sqrt(128)
  const int jmax = q0 >> 4;
  for (int j = 0; j <= jmax; j++) {
    const int c0 = j * 16;
    // ---- S = Q Kt
    v8f sc = vz;
    #pragma unroll 8
    for (int kk = 0; kk < 32; kk++) {
      v2f b = *(const v2f*)(K + (size_t)(c0 + l16) * QKV_STRIDE + kk * 4 + half * 2);
      sc = wmma4(qf[kk], b, sc);
    }
    // ---- online softmax per row
    float alpha[8];
    #pragma unroll
    for (int r = 0; r < 8; r++) {
      const int rowl = half ? 8 + r : r;
      float sv = sc[r] * scale;
      if (c0 + l16 > q0 + rowl) sv = -3.0e38f;        // causal mask
      float mx = sv;
      #pragma unroll
      for (int off = 1; off < 16; off <<= 1) mx = fmaxf(mx, __shfl_xor(mx, off, 16));
      const float mnew = fmaxf(m[r], mx);
      const float a_ = __expf(m[r] - mnew);
      const float p = __expf(sv - mnew);
      float rs = p;
      #pragma unroll
      for (int off = 1; off < 16; off <<= 1) rs += __shfl_xor(rs, off, 16);
      l[r] = l[r] * a_ + rs;
      m[r] = mnew;
      alpha[r] = a_;
      pw[rowl * 18 + l16] = p;
    }
    __asm__ volatile("s_wait_dscnt 0" ::: "memory");
    #pragma unroll
    for (int nn = 0; nn < 8; nn++) {
      #pragma unroll
      for (int r = 0; r < 8; r++) o[nn][r] *= alpha[r];
    }
    // ---- O += P V
    #pragma unroll
    for (int nn = 0; nn < 8; nn++) {
      v8f acc = o[nn];
      #pragma unroll
      for (int k0 = 0; k0 < 16; k0 += 4) {
        const int ka = k0 + half * 2;
        v2f a = *(const v2f*)(pw + l16 * 18 + ka);
        v2f b;
        b.x = V[(size_t)(c0 + ka) * QKV_STRIDE + nn * 16 + l16];
        b.y = V[(size_t)(c0 + ka + 1) * QKV_STRIDE + nn * 16 + l16];
        acc = wmma4(a, b, acc);
      }
      o[nn] = acc;
    }
  }
  #pragma unroll
  for (int nn = 0; nn < 8; nn++) {
    #pragma unroll
    for (int r = 0; r < 8; r++) {
      const int row = q0 + (half ? 8 + r : r);
      xattn[(size_t)row * D_DIM + head * HD + nn * 16 + l16] = o[nn][r] / l[r];
    }
  }
}

// ---------------------------------------------------------------- scatter chunks into O-MoE activation
__global__ void scatter_o_kernel(const float* __restrict__ xattn, const float* __restrict__ g_o,
                                 const int* __restrict__ idx_o, float* __restrict__ A) {
  const int t = blockIdx.x * 256 + threadIdx.x;       // S*128
  const int c = t & 127;
  const int s = t >> 7;
  float* arow = A + (size_t)s * (E_O * HD);
  #pragma unroll
  for (int n = 0; n < NOSLOT; n++) {
    const float v = xattn[(size_t)s * D_DIM + n * HD + c];
    arow[n * HD + c] += v;                            // dense expert n (DLID=0)
    const int e = idx_o[s * NOSLOT + n];
    arow[e * HD + c] += g_o[s * NOSLOT + n] * v;      // routed expert
  }
}

// ---------------------------------------------------------------- xres = x + A[S,6144] @ Opool[6144,768]
__global__ __launch_bounds__(256) void gemm_o_kernel(const float* __restrict__ A,
                                                     const float* __restrict__ B,
                                                     const float* __restrict__ x,
                                                     float* __restrict__ xres) {
  const int lane = threadIdx.x & 31;
  const int wid = blockIdx.x * 8 + (threadIdx.x >> 5);
  const int mt = wid / 24, nt = wid % 24;             // 64 x 24 tiles of 32x32
  const int m0 = mt * 32, n0 = nt * 32;
  const int Kdim = E_O * HD;                          // 6144
  v8f c[2][2];
  zero22(c);
  wmma_tile_rr_2x2(A + (size_t)m0 * Kdim, Kdim, B + n0, D_DIM, Kdim, c, lane);
  const int half = lane >> 4, l16 = lane & 15;
  #pragma unroll
  for (int mi = 0; mi < 2; mi++)
    #pragma unroll
    for (int r = 0; r < 8; r++) {
      const int row = m0 + mi * 16 + (half ? 8 + r : r);
      #pragma unroll
      for (int ni = 0; ni < 2; ni++) {
        const size_t off = (size_t)row * D_DIM + n0 + ni * 16 + l16;
        xres[off] = x[off] + c[mi][ni][r];
      }
    }
}

// ---------------------------------------------------------------- top-4 FFN routing -> comb_f[S,48]
__global__ void route_ffn_kernel(const float* __restrict__ h2, const float* __restrict__ kp,
                                 float* __restrict__ comb) {
  __shared__ float skp[100 * E_FFN];
  for (int i = threadIdx.x; i < 100 * E_FFN; i += 256) skp[i] = kp[i];
  __syncthreads();
  const int s = blockIdx.x * 256 + threadIdx.x;
  if (s >= S_LEN) return;
  const float* q = h2 + (size_t)s * D_DIM;
  float qv[96];
  #pragma unroll
  for (int j = 0; j < 96; j++) qv[j] = q[j];
  float bv[4] = {-3.0e38f, -3.0e38f, -3.0e38f, -3.0e38f};
  int bi[4] = {0, 0, 0, 0};
  for (int e = 0; e < E_FFN; e++) {
    float acc = skp[96 * E_FFN + e];                  // kp[-4]
    #pragma unroll
    for (int j = 0; j < 80; j++) acc += qv[j] * skp[j * E_FFN + e];
    #pragma unroll
    for (int k = 0; k < 4; k++) {
      if (acc > bv[k]) {
        for (int q2 = 3; q2 > k; q2--) { bv[q2] = bv[q2 - 1]; bi[q2] = bi[q2 - 1]; }
        bv[k] = acc; bi[k] = e;
        break;
      }
    }
  }
  float cb[E_FFN];
  for (int e = 0; e < E_FFN; e++) cb[e] = 0.f;
  #pragma unroll
  for (int k = 0; k < 4; k++) {
    const int e = bi[k];
    float gate = 0.f;
    #pragma unroll
    for (int j = 80; j < 96; j++) gate += qv[j] * skp[j * E_FFN + e];
    const float z = bv[k] + gate + skp[97 * E_FFN + e];
    const float sg = 1.f / (1.f + __expf(-z));
    cb[e] += sg * skp[98 * E_FFN + e] + skp[99 * E_FFN + e];
  }
  #pragma unroll
  for (int e = 0; e < 6; e++) cb[e] += 1.0f;          // dense experts, DLID=0
  for (int e = 0; e < E_FFN; e++) comb[(size_t)s * E_FFN + e] = cb[e];
}

// ---------------------------------------------------------------- fused FFN up/gate: hid = silu(u)*w*comb
__global__ __launch_bounds__(256) void ffn_up_kernel(const float* __restrict__ h2,
                                                     const float* __restrict__ pool,
                                                     const float* __restrict__ comb,
                                                     float* __restrict__ hid) {
  const int lane = threadIdx.x & 31;
  const int wid = blockIdx.x * 8 + (threadIdx.x >> 5);
  const int e = wid >> 9;                             // 512 32x32-tiles/expert (64 M x 8 N)
  const int t = wid & 511;
  const int mt = t >> 3, nt = t & 7;
  const int m0 = mt * 32, n0 = nt * 32;
  const int half = lane >> 4, l16 = lane & 15;
  const int KH = E_FFN * FFN_ED;                      // 12288
  float* dst = hid + (size_t)m0 * KH + e * FFN_ED + n0 + l16;
  const float c0v = comb[(size_t)(m0 + l16) * E_FFN + e];
  const float c1v = comb[(size_t)(m0 + 16 + l16) * E_FFN + e];
  if (!__any((c0v != 0.0f) || (c1v != 0.0f))) {
    #pragma unroll
    for (int mi = 0; mi < 2; mi++)
      #pragma unroll
      for (int r = 0; r < 8; r++) {
        const int row = mi * 16 + (half ? 8 + r : r);
        #pragma unroll
        for (int ni = 0; ni < 2; ni++) dst[(size_t)row * KH + ni * 16] = 0.0f;
      }
    return;
  }
  const float* a0 = h2 + (size_t)(m0 + l16) * D_DIM + half * 2;
  const float* a1 = a0 + 16 * D_DIM;
  const float* bu0 = pool + (size_t)(e * 3 + 0) * (D_DIM * FFN_ED) + n0 + l16 + (half * 2) * FFN_ED;
  const float* bu1 = bu0 + 16;
  const float* bw0 = bu0 + D_DIM * FFN_ED;
  const float* bw1 = bw0 + 16;
  v8f cu[2][2], cw[2][2];
  zero22(cu); zero22(cw);
  for (int k0 = 0; k0 < D_DIM; k0 += 4) {
    v2f av0 = *(const v2f*)(a0 + k0);
    v2f av1 = *(const v2f*)(a1 + k0);
    v2f u0, u1, w0, w1;
    u0.x = bu0[k0 * FFN_ED]; u0.y = bu0[k0 * FFN_ED + FFN_ED];
    u1.x = bu1[k0 * FFN_ED]; u1.y = bu1[k0 * FFN_ED + FFN_ED];
    w0.x = bw0[k0 * FFN_ED]; w0.y = bw0[k0 * FFN_ED + FFN_ED];
    w1.x = bw1[k0 * FFN_ED]; w1.y = bw1[k0 * FFN_ED + FFN_ED];
    cu[0][0] = wmma4(av0, u0, cu[0][0]);
    cu[0][1] = wmma4(av0, u1, cu[0][1]);
    cu[1][0] = wmma4(av1, u0, cu[1][0]);
    cu[1][1] = wmma4(av1, u1, cu[1][1]);
    cw[0][0] = wmma4(av0, w0, cw[0][0]);
    cw[0][1] = wmma4(av0, w1, cw[0][1]);
    cw[1][0] = wmma4(av1, w0, cw[1][0]);
    cw[1][1] = wmma4(av1, w1, cw[1][1]);
  }
  #pragma unroll
  for (int mi = 0; mi < 2; mi++)
    #pragma unroll
    for (int r = 0; r < 8; r++) {
      const int rowl = mi * 16 + (half ? 8 + r : r);
      const float cc = comb[(size_t)(m0 + rowl) * E_FFN + e];
      #pragma unroll
      for (int ni = 0; ni < 2; ni++) {
        const float u = cu[mi][ni][r], w = cw[mi][ni][r];
        const float sil = u * (1.0f / (1.0f + __expf(-u)));
        dst[(size_t)rowl * KH + ni * 16] = sil * w * cc;
      }
    }
}

// ---------------------------------------------------------------- out = xres + hid[S,12288] @ V^T
__global__ __launch_bounds__(256) void ffn_down_kernel(const float* __restrict__ hid,
                                                       const float* __restrict__ pool,
                                                       const float* __restrict__ xres,
                                                       float* __restrict__ out) {
  const int lane = threadIdx.x & 31;
  const int wid = blockIdx.x * 8 + (threadIdx.x >> 5);
  const int mt = wid / 24, nt = wid % 24;             // 64 x 24 tiles of 32x32
  const int m0 = mt * 32, n0 = nt * 32;
  const int KH = E_FFN * FFN_ED;                      // 12288
  v8f c[2][2];
  zero22(c);
  for (int e = 0; e < E_FFN; e++) {
    const float* A_e = hid + (size_t)m0 * KH + e * FFN_ED;
    const float* Bt_e = pool + (size_t)(e * 3 + 2) * (D_DIM * FFN_ED) + (size_t)n0 * FFN_ED;
    wmma_tile_rt_2x2(A_e, KH, Bt_e, FFN_ED, FFN_ED, c, lane);
  }
  const int half = lane >> 4, l16 = lane & 15;
  #pragma unroll
  for (int mi = 0; mi < 2; mi++)
    #pragma unroll
    for (int r = 0; r < 8; r++) {
      const int row = m0 + mi * 16 + (half ? 8 + r : r);
      #pragma unroll
      for (int ni = 0; ni < 2; ni++) {
        const size_t off = (size_t)row * D_DIM + n0 + ni * 16 + l16;
        out[off] = xres[off] + c[mi][ni][r];
      }
    }
}

// ================================================================ host side
extern "C" void kernel_launch(void* const* d_in, const int* in_sizes, int n_in,
                              void* d_out, int out_size, void* d_ws, size_t ws_size,
                              hipStream_t stream) {
  (void)in_sizes; (void)n_in; (void)out_size; (void)ws_size;
  const float* x          = (const float*)d_in[0];
  const float* attn_pool  = (const float*)d_in[1];
  const float* attn_o_pool= (const float*)d_in[2];
  const float* ffn_pool   = (const float*)d_in[3];
  const float* akp        = (const float*)d_in[4];
  const float* okp        = (const float*)d_in[5];
  const float* fkp        = (const float*)d_in[6];
  const float* wn1        = (const float*)d_in[7];
  const float* wn2        = (const float*)d_in[8];
  float* out = (float*)d_out;

  char* ws = (char*)d_ws;
  size_t off = 0;
  auto alloc = [&](size_t bytes) -> void* {
    off = (off + 255) & ~(size_t)255;
    void* p = ws + off;
    off += bytes;
    return p;
  };
  float* h      = (float*)alloc((size_t)S_LEN * D_DIM * 4);                 // also h2
  float* allout = (float*)alloc((size_t)S_LEN * E_ATTN * HD * 4);           // also hid
  float* qkv    = (float*)alloc((size_t)S_LEN * NSLOT * HD * 4);
  float* xattn  = (float*)alloc((size_t)S_LEN * D_DIM * 4);
  float* xres   = (float*)alloc((size_t)S_LEN * D_DIM * 4);
  float* Ao     = (float*)alloc((size_t)S_LEN * E_O * HD * 4);
  float* g_attn = (float*)alloc((size_t)S_LEN * NSLOT * 4);
  int*   i_attn = (int*)  alloc((size_t)S_LEN * NSLOT * 4);
  float* g_o    = (float*)alloc((size_t)S_LEN * NOSLOT * 4);
  int*   i_o    = (int*)  alloc((size_t)S_LEN * NOSLOT * 4);
  float* comb   = (float*)alloc((size_t)S_LEN * E_FFN * 4);

  // 1. h = rmsnorm(x, wn1)
  rmsnorm_kernel<<<S_LEN, 256, 0, stream>>>(x, wn1, h);
  // 2. attention routing (top-1 over 96 experts)
  route_top1_kernel<48, 64, E_ATTN><<<(S_LEN * NSLOT + 255) / 256, 256, 0, stream>>>(
      h, akp, g_attn, i_attn, S_LEN * NSLOT, NSLOT, D_DIM, 64);
  // 3. all_out[s,e,:] = h @ attn_pool[e]  (96 per-expert GEMMs, 32x32 WMMA tiles)
  expert_gemm_attn_kernel<<<(E_ATTN * 256) / 8, 256, 0, stream>>>(h, attn_pool, allout);
  // 4. qkv combine
  qkv_build_kernel<<<(S_LEN * NSLOT * HD) / 256, 256, 0, stream>>>(allout, g_attn, i_attn, qkv);
  // 5. RoPE on q,k slots
  rope_kernel<<<(S_LEN * 9 * 64) / 256, 256, 0, stream>>>(qkv);
  // 6. flash attention -> xattn
  flash_attn_kernel<<<(QH * 128) / 8, 256, 0, stream>>>(qkv, xattn);
  // 7. O routing (top-1 over 48 experts, query = first 64 of each 128-chunk)
  route_top1_kernel<48, 64, E_O><<<(S_LEN * NOSLOT + 255) / 256, 256, 0, stream>>>(
      xattn, okp, g_o, i_o, S_LEN * NOSLOT, NOSLOT, D_DIM, HD);
  // 8-9. build sparse O activation
  hipMemsetAsync(Ao, 0, (size_t)S_LEN * E_O * HD * 4, stream);
  scatter_o_kernel<<<(S_LEN * HD) / 256, 256, 0, stream>>>(xattn, g_o, i_o, Ao);
  // 10. xres = x + Ao @ attn_o_pool
  gemm_o_kernel<<<(64 * 24) / 8, 256, 0, stream>>>(Ao, attn_o_pool, x, xres);
  // 11. h2 = rmsnorm(xres, wn2)  (reuses h buffer)
  rmsnorm_kernel<<<S_LEN, 256, 0, stream>>>(xres, wn2, h);
  // 12. FFN routing (top-4) -> comb_f
  route_ffn_kernel<<<(S_LEN + 255) / 256, 256, 0, stream>>>(h, fkp, comb);
  // 13. hid = silu(h2@U)*(h2@W)*comb  (reuses allout buffer)
  ffn_up_kernel<<<(E_FFN * 512) / 8, 256, 0, stream>>>(h, ffn_pool, comb, allout);
  // 14. out = xres + hid @ V^T
  ffn_down_kernel<<<(64 * 24) / 8, 256, 0, stream>>>(allout, ffn_pool, xres, out);
}